// GeometricEncoder_75694503625220
// MI455X (gfx1250) — compile-verified
//
#include <hip/hip_runtime.h>
#include <math.h>

#define NB    4
#define NPTS  8192
#define KNN   16
#define IN_DIM 118
#define FROW  120      // padded feature row (multiple of 4 for K-stepping)
#define HID   64

typedef __attribute__((ext_vector_type(2))) float v2f;
typedef __attribute__((ext_vector_type(8))) float v8f;
typedef unsigned long long u64;

static __device__ __forceinline__ u64 umin64(u64 a, u64 b) { return a < b ? a : b; }
static __device__ __forceinline__ u64 umax64(u64 a, u64 b) { return a > b ? a : b; }

// =====================================================================
// Kernel 1: brute-force KNN via V_WMMA_F32_16X16X4_F32
//   D[m][n] = (cand_m . -2 q_n)  +  (|c_m|^2 + |q_n|^2)   ==  d^2
// Queries sit on the N axis so each lane's 8 D-values share one query.
// Top-16 kept as packed u64 keys (d2_bits<<32 | idx) in registers and
// maintained with a branchless min/max insertion network; half-lane
// lists merged through LDS at the end.
// =====================================================================
#define CHUNK 1024     // candidate points staged in LDS per pass (16 KB)
#define QPB   128      // queries per block = 8 waves * 16

__global__ __launch_bounds__(256) void knn_kernel(const float* __restrict__ x,
                                                  int* __restrict__ idx_out) {
  __shared__ float4 cand[CHUNK];        // 16 KB  (xyz + |c|^2)
  __shared__ u64    mk[8 * 16 * 32];    // 32 KB  merge keys

  const int tid = threadIdx.x;
  const int w   = tid >> 5;             // wave id 0..7
  const int l   = tid & 31;             // lane within wave (wave32)
  const int h   = l >> 4;               // half-wave (0/1)
  const int ln  = l & 15;

  const int b   = blockIdx.x / (NPTS / QPB);
  const int q0b = (blockIdx.x % (NPTS / QPB)) * QPB;
  const int q0  = q0b + w * 16;         // this wave's first query
  const int qg  = q0 + ln;              // this lane's query (in-batch index)

  const float* xb = x + (size_t)b * NPTS * 3;

  // Query point, |q|^2, and the B fragment (constant for the whole kernel).
  const float qx = xb[qg * 3 + 0];
  const float qy = xb[qg * 3 + 1];
  const float qz = xb[qg * 3 + 2];
  const float qsq = qx * qx + qy * qy + qz * qz;
  v2f bq;
  bq[0] = h ? (-2.0f * qz) : (-2.0f * qx);
  bq[1] = h ? 0.0f         : (-2.0f * qy);

  u64 best[KNN];
#pragma unroll
  for (int j = 0; j < KNN; ++j) best[j] = ~0ull;

  for (int c0 = 0; c0 < NPTS; c0 += CHUNK) {
    __syncthreads();
    // Cooperative stage of CHUNK candidates into LDS (coalesced).
    for (int j = tid; j < CHUNK; j += 256) {
      const int g = c0 + j;
      const float cx = xb[g * 3 + 0];
      const float cy = xb[g * 3 + 1];
      const float cz = xb[g * 3 + 2];
      cand[j] = make_float4(cx, cy, cz, cx * cx + cy * cy + cz * cz);
    }
    if (c0 + CHUNK < NPTS)
      __builtin_prefetch(&xb[(size_t)(c0 + CHUNK) * 3], 0, 1);  // global_prefetch
    __syncthreads();

#pragma unroll 2
    for (int t = 0; t < CHUNK / 16; ++t) {
      const int tb = t * 16;
      const float4 cp = cand[tb + ln];
      v2f a;
      a[0] = h ? cp.z : cp.x;
      a[1] = h ? 0.0f : cp.y;
      v8f c;
#pragma unroll
      for (int r = 0; r < 8; ++r)
        c[r] = cand[tb + r + 8 * h].w + qsq;      // |c_m|^2 + |q_n|^2

      v8f d2 = __builtin_amdgcn_wmma_f32_16x16x4_f32(
          false, a, false, bq, (short)0, c, false, false);

      const int cbase = c0 + tb + 8 * h;
#pragma unroll
      for (int r = 0; r < 8; ++r) {
        const int cg = cbase + r;
        // Non-negative d2 -> bit pattern is order-preserving; low word is
        // the index, so ties break by lower index (stable, like top_k).
        const float dv = fmaxf(d2[r], 0.0f);
        u64 key = ((u64)__float_as_uint(dv) << 32) | (unsigned)cg;
        key = (cg == qg) ? ~0ull : key;           // exclude diagonal
        if (key < best[KNN - 1]) {                // rare: enters top-16
          // Branchless min/max insertion network (in-place, descending j).
#pragma unroll
          for (int j = KNN - 1; j >= 1; --j)
            best[j] = umax64(best[j - 1], umin64(best[j], key));
          best[0] = umin64(best[0], key);
        }
      }
    }
  }

  // Each half-lane pair (l, l+16) holds a sorted top-16 over disjoint halves
  // of the candidate stream; merge the two sorted lists through LDS.
  {
    const int base = (w * 16 + ln) * 32 + h * 16;
#pragma unroll
    for (int j = 0; j < KNN; ++j) mk[base + j] = best[j];
  }
  __syncthreads();

  if (l < 16) {
    const int p0 = (w * 16 + l) * 32;
    int i0 = 0, i1 = 0;
    int* dst = idx_out + ((size_t)b * NPTS + (q0 + l)) * KNN;
#pragma unroll
    for (int k = 0; k < KNN; ++k) {
      const u64 k0v = (i0 < 16) ? mk[p0 + i0]      : ~0ull;
      const u64 k1v = (i1 < 16) ? mk[p0 + 16 + i1] : ~0ull;
      int pick;
      if (k0v <= k1v) { pick = (int)(unsigned)k0v; ++i0; }
      else            { pick = (int)(unsigned)k1v; ++i1; }
      dst[k] = pick;
    }
  }
}

// =====================================================================
// Kernel 2: features (gather, covariance, closed-form 3x3 eigenvalues)
// + MLP. Layer 1 (118->64) runs as WMMA f32 16x16x4 with b1 folded into
// the C operand; layer 2 (64->3) is trivially VALU.
// Block = 128 threads (4 waves), 64 points per block.
// =====================================================================
#define PPB 64

__global__ __launch_bounds__(128) void encode_kernel(const float* __restrict__ x,
                                                     const float* __restrict__ W1,
                                                     const float* __restrict__ b1,
                                                     const float* __restrict__ W2,
                                                     const float* __restrict__ b2,
                                                     const int* __restrict__ idx,
                                                     float* __restrict__ out) {
  __shared__ float feats[PPB * FROW];   // 30720 B, reused as h-buffer after WMMA
  __shared__ float w1sh[FROW * HID];    // 30720 B, rows 118..119 zero-padded

  const int tid = threadIdx.x;
  const int w  = tid >> 5;
  const int l  = tid & 31;
  const int h  = l >> 4;
  const int ln = l & 15;

  // Stage W1 into LDS with zero K-padding.
  for (int i = tid; i < FROW * HID; i += 128)
    w1sh[i] = (i < IN_DIM * HID) ? W1[i] : 0.0f;

  const int p0 = blockIdx.x * PPB;      // flat point id over B*N

  // ---- Phase 1: one thread per point builds the 118-dim feature row ----
  if (tid < PPB) {
    const int pg = p0 + tid;
    const int bb = pg / NPTS;
    const int qn = pg - bb * NPTS;
    const float* xb = x + (size_t)bb * NPTS * 3;
    const int* nb = idx + (size_t)pg * KNN;

    const float px = xb[qn * 3 + 0];
    const float py = xb[qn * 3 + 1];
    const float pz = xb[qn * 3 + 2];
    float* f = feats + tid * FROW;
    f[0] = px; f[1] = py; f[2] = pz;

    float c00 = 0.f, c01 = 0.f, c02 = 0.f, c11 = 0.f, c12 = 0.f, c22 = 0.f;
#pragma unroll
    for (int k = 0; k < KNN; ++k) {
      const int j = nb[k];
      const float nx = xb[j * 3 + 0];
      const float ny = xb[j * 3 + 1];
      const float nz = xb[j * 3 + 2];
      const float rx = nx - px, ry = ny - py, rz = nz - pz;
      f[3  + k * 3 + 0] = nx; f[3  + k * 3 + 1] = ny; f[3  + k * 3 + 2] = nz;
      f[51 + k * 3 + 0] = rx; f[51 + k * 3 + 1] = ry; f[51 + k * 3 + 2] = rz;
      f[99 + k] = sqrtf(rx * rx + ry * ry + rz * rz);
      c00 += rx * rx; c01 += rx * ry; c02 += rx * rz;
      c11 += ry * ry; c12 += ry * rz; c22 += rz * rz;
    }
    const float inv = 1.0f / ((float)(KNN - 1) + 1e-6f);
    c00 *= inv; c01 *= inv; c02 *= inv; c11 *= inv; c12 *= inv; c22 *= inv;

    // Closed-form symmetric 3x3 eigenvalues (descending l1 >= l2 >= l3).
    const float q  = (c00 + c11 + c22) * (1.0f / 3.0f);
    const float p1 = c01 * c01 + c02 * c02 + c12 * c12;
    const float a0 = c00 - q, a1 = c11 - q, a2 = c22 - q;
    const float p2 = a0 * a0 + a1 * a1 + a2 * a2 + 2.0f * p1;
    const float p  = sqrtf(p2 * (1.0f / 6.0f));
    float l1, l2, l3;
    if (p < 1e-12f) {
      l1 = l2 = l3 = q;
    } else {
      const float ip = 1.0f / p;
      const float b00 = a0 * ip, b11 = a1 * ip, b22 = a2 * ip;
      const float b01 = c01 * ip, b02 = c02 * ip, b12 = c12 * ip;
      float r = 0.5f * (b00 * (b11 * b22 - b12 * b12)
                      - b01 * (b01 * b22 - b12 * b02)
                      + b02 * (b01 * b12 - b11 * b02));
      r = fminf(1.0f, fmaxf(-1.0f, r));
      const float phi = acosf(r) * (1.0f / 3.0f);
      l1 = q + 2.0f * p * __cosf(phi);
      l3 = q + 2.0f * p * __cosf(phi + 2.0943951023931953f);
      l2 = 3.0f * q - l1 - l3;
    }
    const float den = 1.0f / (l1 + 1e-6f);
    f[115] = (l1 - l2) * den;
    f[116] = (l2 - l3) * den;
    f[117] = l3 * den;
    f[118] = 0.0f; f[119] = 0.0f;       // K padding
  }
  __syncthreads();

  // ---- Phase 2: layer 1 via WMMA f32 16x16x4 (16 points x 64 hidden) ----
  v8f acc0, acc1, acc2, acc3;
  {
    const float vb0 = b1[ 0 + ln], vb1 = b1[16 + ln];
    const float vb2 = b1[32 + ln], vb3 = b1[48 + ln];
#pragma unroll
    for (int r = 0; r < 8; ++r) { acc0[r] = vb0; acc1[r] = vb1; acc2[r] = vb2; acc3[r] = vb3; }
  }
  const int arow = (16 * w + ln) * FROW + 2 * h;
#pragma unroll
  for (int k0 = 0; k0 < FROW; k0 += 4) {
    v2f a;
    a[0] = feats[arow + k0];
    a[1] = feats[arow + k0 + 1];
    const int kA = k0 + 2 * h;
    const float* wr0 = &w1sh[kA * HID + ln];
    const float* wr1 = &w1sh[(kA + 1) * HID + ln];
    v2f fb0; fb0[0] = wr0[ 0]; fb0[1] = wr1[ 0];
    v2f fb1; fb1[0] = wr0[16]; fb1[1] = wr1[16];
    v2f fb2; fb2[0] = wr0[32]; fb2[1] = wr1[32];
    v2f fb3; fb3[0] = wr0[48]; fb3[1] = wr1[48];
    acc0 = __builtin_amdgcn_wmma_f32_16x16x4_f32(false, a, false, fb0, (short)0, acc0, false, false);
    acc1 = __builtin_amdgcn_wmma_f32_16x16x4_f32(false, a, false, fb1, (short)0, acc1, false, false);
    acc2 = __builtin_amdgcn_wmma_f32_16x16x4_f32(false, a, false, fb2, (short)0, acc2, false, false);
    acc3 = __builtin_amdgcn_wmma_f32_16x16x4_f32(false, a, false, fb3, (short)0, acc3, false, false);
  }

  // ReLU + store h into the (now consumed) feats buffer.
#pragma unroll
  for (int r = 0; r < 8; ++r) {
    const int row = 16 * w + r + 8 * h;
    float* hp = feats + row * FROW;
    hp[ 0 + ln] = fmaxf(acc0[r], 0.0f);
    hp[16 + ln] = fmaxf(acc1[r], 0.0f);
    hp[32 + ln] = fmaxf(acc2[r], 0.0f);
    hp[48 + ln] = fmaxf(acc3[r], 0.0f);
  }
  __syncthreads();

  // ---- Phase 3: layer 2 (64->3) + ReLU, one lane per point ----
  if (l < 16) {
    const int row = 16 * w + l;
    const float* hp = feats + row * FROW;
    float o0 = b2[0], o1 = b2[1], o2 = b2[2];
#pragma unroll
    for (int j = 0; j < HID; ++j) {
      const float hv = hp[j];
      o0 += hv * W2[j * 3 + 0];
      o1 += hv * W2[j * 3 + 1];
      o2 += hv * W2[j * 3 + 2];
    }
    const size_t og = (size_t)(p0 + row) * 3;
    out[og + 0] = fmaxf(o0, 0.0f);
    out[og + 1] = fmaxf(o1, 0.0f);
    out[og + 2] = fmaxf(o2, 0.0f);
  }
}

extern "C" void kernel_launch(void* const* d_in, const int* in_sizes, int n_in,
                              void* d_out, int out_size, void* d_ws, size_t ws_size,
                              hipStream_t stream) {
  const float* x  = (const float*)d_in[0];
  const float* W1 = (const float*)d_in[1];
  const float* b1 = (const float*)d_in[2];
  const float* W2 = (const float*)d_in[3];
  const float* b2 = (const float*)d_in[4];
  float* out = (float*)d_out;
  int* idx = (int*)d_ws;                 // B*N*16 ints = 2 MB scratch

  knn_kernel<<<NB * (NPTS / QPB), 256, 0, stream>>>(x, idx);
  encode_kernel<<<(NB * NPTS) / PPB, 128, 0, stream>>>(x, W1, b1, W2, b2, idx, out);
}